// HopfBifurCpxRNNLayer_80221399154908
// MI455X (gfx1250) — compile-verified
//
#include <hip/hip_runtime.h>
#include <math.h>

// ---------------------------------------------------------------------------
// HopfBifurCpxRNNLayer for MI455X (gfx1250, wave32).
//   out[a,b,t,r] = tanh( Σ_s ism[a,t,s]·G[b,s,r] + ms[a,b,r] )
//   ism[b,t,s]   = Σ_d (x@M)[b,t,d] · exp(-i·angle(st[b,s])·(d+1))
// Complex f32 GEMMs through V_WMMA_F32_16X16X4_F32 with:
//   * Karatsuba 3-multiply complex product (25% fewer WMMA issues)
//   * 128x32 block tiles, B strip staged in LDS (shared by 8 waves)
//   * 16x32 per-wave output (A fragment reused across 2 N-tiles)
// ---------------------------------------------------------------------------

typedef float v2f __attribute__((ext_vector_type(2)));
typedef float v8f __attribute__((ext_vector_type(8)));

#define WMMA_F32(a, b, c) \
  __builtin_amdgcn_wmma_f32_16x16x4_f32(false, (a), false, (b), (short)0, (c), false, false)

constexpr int BB = 8, TT = 1024, DD = 256, SS = 256;
constexpr int KBLK = 32;   // K-block staged in LDS per barrier
constexpr int BSTR = 40;   // padded LDS row stride (floats): 2*BSTR % 64 == 16
                           // -> half-wave row pairs hit disjoint bank sets

// -------------------- sprojT[b,d,s] = exp(-i*angle(st[b,s])*(d+1)) ----------
__global__ void __launch_bounds__(256)
sproj_kernel(const float* __restrict__ st_re, const float* __restrict__ st_im,
             float* __restrict__ spT_re, float* __restrict__ spT_im) {
  int i = blockIdx.x * blockDim.x + threadIdx.x;   // over B*D*S, layout [b][d][s]
  int b   = i / (DD * SS);
  int rem = i % (DD * SS);
  int d   = rem / SS;
  int s   = rem % SS;
  float ang = atan2f(st_im[b * SS + s], st_re[b * SS + s]);
  float ph  = ang * (float)(d + 1);
  float sn, cs;
  sincosf(ph, &sn, &cs);
  spT_re[i] = cs;     // Re exp(-i*ph) =  cos(ph)
  spT_im[i] = -sn;    // Im exp(-i*ph) = -sin(ph)
}

// -------------------- ms[a*B+b, r] = Σ_s st[a,s]*H[b,s,r] ------------------
__global__ void __launch_bounds__(256)
ms_kernel(const float* __restrict__ st_re, const float* __restrict__ st_im,
          const float* __restrict__ H_re,  const float* __restrict__ H_im,
          float* __restrict__ ms_re, float* __restrict__ ms_im) {
  int i = blockIdx.x * blockDim.x + threadIdx.x;   // over B*B*S
  int r = i % SS;
  int p = i / SS;
  int a = p / BB;
  int b = p % BB;
  float accr = 0.f, acci = 0.f;
  for (int s = 0; s < SS; ++s) {
    float xr = st_re[a * SS + s], xi = st_im[a * SS + s];
    float hr = H_re[((size_t)b * SS + s) * SS + r];
    float hi = H_im[((size_t)b * SS + s) * SS + r];
    accr += xr * hr - xi * hi;
    acci += xr * hi + xi * hr;
  }
  ms_re[i] = accr;
  ms_im[i] = acci;
}

// ---- core complex-GEMM block body (shared by both GEMM kernels) ------------
// Computes the 6 Karatsuba accumulators for a 16x32 per-wave tile.
// A planar (row-major, stride K), B strip staged in LDS (KBLK x 32, stride BSTR).
struct CAcc { v8f p1a, p2a, p3a, p1b, p2b, p3b; };

__device__ __forceinline__ void
cgemm_block(CAcc& acc, const float* __restrict__ pAre, const float* __restrict__ pAim,
            const float* __restrict__ sBre, const float* __restrict__ sBim,
            int kb, int half, int lr) {
#pragma unroll
  for (int ks = 0; ks < KBLK; ks += 4) {
    const int ak = kb + ks + half * 2;   // A layout: lanes 16-31 carry K+2,K+3
    const int lk = ks + half * 2;        // B layout: VGPR0 rows K / K+2
    v2f ar, ai;
    float2 t;
    t = *reinterpret_cast<const float2*>(pAre + ak); ar[0] = t.x; ar[1] = t.y;
    t = *reinterpret_cast<const float2*>(pAim + ak); ai[0] = t.x; ai[1] = t.y;
    const v2f as = ar + ai;

    v2f br0, bi0, br1, bi1;
    br0[0] = sBre[lk * BSTR + lr];        br0[1] = sBre[(lk + 1) * BSTR + lr];
    bi0[0] = sBim[lk * BSTR + lr];        bi0[1] = sBim[(lk + 1) * BSTR + lr];
    br1[0] = sBre[lk * BSTR + lr + 16];   br1[1] = sBre[(lk + 1) * BSTR + lr + 16];
    bi1[0] = sBim[lk * BSTR + lr + 16];   bi1[1] = sBim[(lk + 1) * BSTR + lr + 16];
    const v2f bs0 = br0 + bi0;
    const v2f bs1 = br1 + bi1;

    acc.p1a = WMMA_F32(ar, br0, acc.p1a);
    acc.p2a = WMMA_F32(ai, bi0, acc.p2a);
    acc.p3a = WMMA_F32(as, bs0, acc.p3a);
    acc.p1b = WMMA_F32(ar, br1, acc.p1b);
    acc.p2b = WMMA_F32(ai, bi1, acc.p2b);
    acc.p3b = WMMA_F32(as, bs1, acc.p3b);
  }
}

// -------------------- batched complex GEMM: C[b] = A[b] @ B[b] --------------
// A (B,M,K), Bm (B,K,N), C (B,M,N), all row-major planar re/im.
// Block (8 waves) -> 128x32 output tile; wave w -> rows w*16..w*16+15.
__global__ void __launch_bounds__(256)
cgemm_batched(const float* __restrict__ Are, const float* __restrict__ Aim,
              const float* __restrict__ Bre, const float* __restrict__ Bim,
              float* __restrict__ Cre, float* __restrict__ Cim,
              int M, int N, int K) {
  __shared__ float sBre[KBLK * BSTR];
  __shared__ float sBim[KBLK * BSTR];

  const int tix  = threadIdx.x;
  const int wave = tix >> 5;
  const int lane = tix & 31;
  const int half = lane >> 4;
  const int lr   = lane & 15;

  const int nbn = N >> 5;                       // 32-col blocks
  const int blocksPerBatch = (M >> 7) * nbn;    // 128-row blocks
  const int b  = blockIdx.x / blocksPerBatch;
  const int bt = blockIdx.x % blocksPerBatch;
  const int bm = bt / nbn;
  const int bn = bt % nbn;

  const int am      = bm * 128 + wave * 16 + lr;
  const int colbase = bn * 32;
  const float* pAre = Are + ((size_t)b * M + am) * K;
  const float* pAim = Aim + ((size_t)b * M + am) * K;
  const float* pBre = Bre + (size_t)b * K * N + colbase;
  const float* pBim = Bim + (size_t)b * K * N + colbase;

  CAcc acc = {};

  for (int kb = 0; kb < K; kb += KBLK) {
    __syncthreads();                            // prior block's LDS reads done
    for (int e = tix; e < KBLK * 32; e += 256) {
      const int kk = e >> 5, cc = e & 31;
      sBre[kk * BSTR + cc] = pBre[(size_t)(kb + kk) * N + cc];
      sBim[kk * BSTR + cc] = pBim[(size_t)(kb + kk) * N + cc];
    }
    __syncthreads();
    cgemm_block(acc, pAre, pAim, sBre, sBim, kb, half, lr);
  }

  const int col0 = colbase + lr;
#pragma unroll
  for (int v = 0; v < 8; ++v) {
    const int row = bm * 128 + wave * 16 + v + half * 8;
    const size_t idx = ((size_t)b * M + row) * N + col0;
    Cre[idx]      = acc.p1a[v] - acc.p2a[v];
    Cim[idx]      = acc.p3a[v] - acc.p1a[v] - acc.p2a[v];
    Cre[idx + 16] = acc.p1b[v] - acc.p2b[v];
    Cim[idx + 16] = acc.p3b[v] - acc.p1b[v] - acc.p2b[v];
  }
}

// ---- final: out[a,b,t,r] = ctanh( ism[a]@G[b] + ms[a,b,:] ), interleaved ---
__device__ __forceinline__ float2 ctanh_f(float re, float im) {
  // tanh(x+iy) = (sinh 2x + i sin 2y) / (cosh 2x + cos 2y), clamp 2x.
  const float x2 = fminf(fmaxf(2.f * re, -80.f), 80.f);
  const float y2 = 2.f * im;
  float sn, cs;
  sincosf(y2, &sn, &cs);
  const float den = coshf(x2) + cs;
  float2 o;
  o.x = sinhf(x2) / den;
  o.y = sn / den;
  return o;
}

__global__ void __launch_bounds__(256)
cgemm_pair_tanh(const float* __restrict__ Are, const float* __restrict__ Aim,  // ism (B,T,S)
                const float* __restrict__ Gre, const float* __restrict__ Gim,  // (B,S,S)
                const float* __restrict__ MSre, const float* __restrict__ MSim,// (B*B,S)
                float* __restrict__ out) {                                     // (B,B,T,S) cpx
  __shared__ float sBre[KBLK * BSTR];
  __shared__ float sBim[KBLK * BSTR];

  const int tix  = threadIdx.x;
  const int wave = tix >> 5;
  const int lane = tix & 31;
  const int half = lane >> 4;
  const int lr   = lane & 15;

  constexpr int nbn = SS / 32;                       // 8
  constexpr int blocksPerPair = (TT / 128) * nbn;    // 64
  const int pair = blockIdx.x / blocksPerPair;
  const int bt   = blockIdx.x % blocksPerPair;
  const int a    = pair / BB;
  const int bb   = pair % BB;
  const int bm   = bt / nbn;
  const int bn   = bt % nbn;

  const int am      = bm * 128 + wave * 16 + lr;
  const int colbase = bn * 32;
  const float* pAre = Are + ((size_t)a * TT + am) * SS;
  const float* pAim = Aim + ((size_t)a * TT + am) * SS;
  const float* pBre = Gre + (size_t)bb * SS * SS + colbase;
  const float* pBim = Gim + (size_t)bb * SS * SS + colbase;

  CAcc acc = {};

  for (int kb = 0; kb < SS; kb += KBLK) {
    __syncthreads();
    for (int e = tix; e < KBLK * 32; e += 256) {
      const int kk = e >> 5, cc = e & 31;
      sBre[kk * BSTR + cc] = pBre[(size_t)(kb + kk) * SS + cc];
      sBim[kk * BSTR + cc] = pBim[(size_t)(kb + kk) * SS + cc];
    }
    __syncthreads();
    cgemm_block(acc, pAre, pAim, sBre, sBim, kb, half, lr);
  }

  const int col0 = colbase + lr;
  const float br0 = MSre[(size_t)pair * SS + col0];
  const float bi0 = MSim[(size_t)pair * SS + col0];
  const float br1 = MSre[(size_t)pair * SS + col0 + 16];
  const float bi1 = MSim[(size_t)pair * SS + col0 + 16];

#pragma unroll
  for (int v = 0; v < 8; ++v) {
    const int row = bm * 128 + wave * 16 + v + half * 8;
    const size_t idx = ((size_t)pair * TT + row) * SS + col0;
    const float re0 = acc.p1a[v] - acc.p2a[v] + br0;
    const float im0 = acc.p3a[v] - acc.p1a[v] - acc.p2a[v] + bi0;
    const float re1 = acc.p1b[v] - acc.p2b[v] + br1;
    const float im1 = acc.p3b[v] - acc.p1b[v] - acc.p2b[v] + bi1;
    *reinterpret_cast<float2*>(out + 2 * idx)        = ctanh_f(re0, im0);
    *reinterpret_cast<float2*>(out + 2 * (idx + 16)) = ctanh_f(re1, im1);
  }
}

// ---------------------------------------------------------------------------
extern "C" void kernel_launch(void* const* d_in, const int* in_sizes, int n_in,
                              void* d_out, int out_size, void* d_ws, size_t ws_size,
                              hipStream_t stream) {
  (void)in_sizes; (void)n_in; (void)out_size; (void)ws_size;
  const float* x_re  = (const float*)d_in[0];
  const float* x_im  = (const float*)d_in[1];
  const float* G_re  = (const float*)d_in[2];
  const float* G_im  = (const float*)d_in[3];
  const float* H_re  = (const float*)d_in[4];
  const float* H_im  = (const float*)d_in[5];
  const float* M_re  = (const float*)d_in[6];
  const float* M_im  = (const float*)d_in[7];
  const float* st_re = (const float*)d_in[8];
  const float* st_im = (const float*)d_in[9];

  float* ws = (float*)d_ws;
  size_t o = 0;
  float* imap_re = ws + o; o += (size_t)BB * TT * DD;
  float* imap_im = ws + o; o += (size_t)BB * TT * DD;
  float* ism_re  = ws + o; o += (size_t)BB * TT * SS;
  float* ism_im  = ws + o; o += (size_t)BB * TT * SS;
  float* spT_re  = ws + o; o += (size_t)BB * DD * SS;
  float* spT_im  = ws + o; o += (size_t)BB * DD * SS;
  float* msv_re  = ws + o; o += (size_t)BB * BB * SS;
  float* msv_im  = ws + o; o += (size_t)BB * BB * SS;

  // 1) phase tensor sprojT (B,D,S) and 2) state bias ms (B*B,S)
  sproj_kernel<<<(BB * DD * SS) / 256, 256, 0, stream>>>(st_re, st_im, spT_re, spT_im);
  ms_kernel<<<(BB * BB * SS) / 256, 256, 0, stream>>>(st_re, st_im, H_re, H_im, msv_re, msv_im);

  // 3) imap = x @ M       : per-batch (T,D)@(D,D)
  cgemm_batched<<<BB * (TT / 128) * (DD / 32), 256, 0, stream>>>(
      x_re, x_im, M_re, M_im, imap_re, imap_im, TT, DD, DD);

  // 4) ism = imap @ sprojT : per-batch (T,D)@(D,S)
  cgemm_batched<<<BB * (TT / 128) * (SS / 32), 256, 0, stream>>>(
      imap_re, imap_im, spT_re, spT_im, ism_re, ism_im, TT, SS, DD);

  // 5) out = tanh(ism[a] @ G[b] + ms[a,b])  over all 64 (a,b) pairs
  cgemm_pair_tanh<<<BB * BB * (TT / 128) * (SS / 32), 256, 0, stream>>>(
      ism_re, ism_im, G_re, G_im, msv_re, msv_im, (float*)d_out);
}